// SoftSplit_40973988004143
// MI455X (gfx1250) — compile-verified
//
#include <hip/hip_runtime.h>
#include <hip/hip_bf16.h>
#include <stdint.h>

typedef __attribute__((ext_vector_type(2))) float v2f;
typedef __attribute__((ext_vector_type(8))) float v8f;

#define NT_   16
#define C_    128
#define H_    120
#define W_    216
#define HID_  512
#define FH_   40
#define FW_   72
#define KHW_  49
#define KDIM_ (C_ * KHW_)       // 6272
#define PIX_  (NT_ * FH_ * FW_) // 46080
#define HW_   (H_ * W_)         // 25920
#define AST_  17                // A LDS row stride (bank spread)
#define BST_  264               // B LDS row stride (1056 B: 16B-aligned, disjoint banks per lane-half)
#define NCH_  392               // 49 taps * 8 channel-chunks of 16

// ---------------------------------------------------------------------------
// One-time weight transpose: W (HID, KDIM) row-major -> Wt (KDIM, HID).
// ---------------------------------------------------------------------------
__global__ __launch_bounds__(256) void wtrans_kernel(const float* __restrict__ W,
                                                     float* __restrict__ Wt) {
    __shared__ float tile[32][33];
    const int k0 = blockIdx.x * 32;
    const int n0 = blockIdx.y * 32;
    const int tx = threadIdx.x;   // 0..31
    const int ty = threadIdx.y;   // 0..7
#pragma unroll
    for (int i = 0; i < 32; i += 8)
        tile[ty + i][tx] = W[(size_t)(n0 + ty + i) * KDIM_ + (k0 + tx)];
    __syncthreads();
#pragma unroll
    for (int i = 0; i < 32; i += 8)
        Wt[(size_t)(k0 + ty + i) * HID_ + (n0 + tx)] = tile[tx][ty + i];
}

// ---------------------------------------------------------------------------
// Implicit-GEMM conv via V_WMMA_F32_16X16X4_F32, double-buffered LDS with
// async global->LDS DMA (ASYNCcnt) for the B (weight) tile.
// Block tile M=64 x N=256; 8 waves, each wave 32x64 (2x4 accumulators).
// ---------------------------------------------------------------------------
__global__ __launch_bounds__(256) void softsplit_gemm(const float* __restrict__ x,
                                                      const float* __restrict__ W,
                                                      const float* __restrict__ Wt,
                                                      const float* __restrict__ bias,
                                                      float* __restrict__ out,
                                                      int useWt) {
    __shared__ float Alds[2][64 * AST_];  // 2 x (64 pixels x 16 k)
    __shared__ float Blds[2][16 * BST_];  // 2 x (16 k x 256 n)

    const int t     = threadIdx.x;
    const int mBase = blockIdx.x * 64;
    const int nBase = blockIdx.y * 256;

    // ---- per-thread A-staging coordinates (one pixel per thread) ----
    const int mLoc = t & 63;
    const int kkA0 = t >> 6;                 // kk = kkA0 + 4u
    const int p    = mBase + mLoc;
    const int nt   = p / (FH_ * FW_);
    const int rem  = p % (FH_ * FW_);
    const int oi   = rem / FW_;
    const int oj   = rem % FW_;
    const int ih0  = oi * 3 - 3;
    const int iw0  = oj * 3 - 3;
    const int xNT  = nt * (C_ * HW_);

    // ---- B async staging coordinates (b128: 16 lanes x 4 x 16B = 1024B rows) ----
    const int bKK  = t >> 4;                 // LDS row 0..15
    const int bN0  = (t & 15) * 16;          // 16 floats per thread (4 x b128)

    // ---- wave / lane mapping ----
    const int wid    = t >> 5;
    const int lane   = t & 31;
    const int laneN  = lane & 15;
    const int laneHi = lane >> 4;            // 0: K0/K1 half, 1: K2/K3 half
    const int waveM  = (wid & 1) * 32;
    const int waveN  = (wid >> 1) * 64;

    v8f acc[2][4];
#pragma unroll
    for (int ms = 0; ms < 2; ++ms)
#pragma unroll
        for (int ns = 0; ns < 4; ++ns)
            acc[ms][ns] = (v8f)0.0f;

    // ---- staging helpers (chunk ch = tap*8 + cchunk) ----
    auto stageA = [&](int ch, int buf) {
        const int tap = ch >> 3;
        const int c0  = (ch & 7) << 4;
        const int ki  = tap / 7;
        const int kj  = tap - ki * 7;
        const int ih  = ih0 + ki;
        const int iw  = iw0 + kj;
        const bool valid = (ih >= 0) && (ih < H_) && (iw >= 0) && (iw < W_);
        const int xbase  = xNT + ih * W_ + iw;
        float* Ab = Alds[buf];
#pragma unroll
        for (int u = 0; u < 4; ++u) {
            const int kk = kkA0 + u * 4;
            float v = 0.0f;
            if (valid) v = x[xbase + (c0 + kk) * HW_];
            Ab[mLoc * AST_ + kk] = v;
        }
    };

    auto stageB = [&](int ch, int buf) {
        const int tap = ch >> 3;
        const int c0  = (ch & 7) << 4;
        const int k   = (c0 + bKK) * KHW_ + tap;
        if (useWt) {
            // async DMA: global -> LDS, 4 x b128 per thread, fully coalesced
            const float* g = Wt + (size_t)k * HID_ + nBase + bN0;
            uint32_t l = (uint32_t)(uintptr_t)&Blds[buf][bKK * BST_ + bN0];
#pragma unroll
            for (int u = 0; u < 4; ++u) {
                asm volatile("global_load_async_to_lds_b128 %0, %1, off"
                             :: "v"(l + (uint32_t)(u * 16)), "v"(g + u * 4)
                             : "memory");
            }
        } else {
            // fallback: direct gather from untransposed weight
            float* Bb = Blds[buf];
#pragma unroll
            for (int u = 0; u < 4; ++u) {
                const int n = bN0 + u * 4;
#pragma unroll
                for (int e = 0; e < 4; ++e)
                    Bb[bKK * BST_ + n + e] =
                        W[(size_t)(nBase + n + e) * KDIM_ + k];
            }
        }
    };

    // ---- software pipeline: prologue stages chunk 0 ----
    stageB(0, 0);
    stageA(0, 0);

    for (int ch = 0; ch < NCH_; ++ch) {
        if (useWt) asm volatile("s_wait_asynccnt 0" ::: "memory");
        __syncthreads();   // chunk ch resident in buf ch&1; buf (ch+1)&1 free

        const int nb = ch + 1;
        if (nb < NCH_) {
            stageB(nb, nb & 1);   // async DMA overlaps the compute below
            stageA(nb, nb & 1);
        }

        const float* Ab = Alds[ch & 1];
        const float* Bb = Blds[ch & 1];
#pragma unroll
        for (int ks = 0; ks < 4; ++ks) {
            const int kc = ks * 4 + laneHi * 2;  // lane-half K base
            v2f a[2], b[4];
#pragma unroll
            for (int ms = 0; ms < 2; ++ms) {
                const int m = waveM + ms * 16 + laneN;
                a[ms].x = Ab[m * AST_ + kc];
                a[ms].y = Ab[m * AST_ + kc + 1];
            }
#pragma unroll
            for (int ns = 0; ns < 4; ++ns) {
                const int n = waveN + ns * 16 + laneN;
                b[ns].x = Bb[kc * BST_ + n];
                b[ns].y = Bb[(kc + 1) * BST_ + n];
            }
#pragma unroll
            for (int ms = 0; ms < 2; ++ms)
#pragma unroll
                for (int ns = 0; ns < 4; ++ns)
                    acc[ms][ns] = __builtin_amdgcn_wmma_f32_16x16x4_f32(
                        false, a[ms], false, b[ns],
                        (short)0, acc[ms][ns], false, false);
        }
    }

    // ---- epilogue: bias add + store (C/D layout: VGPR r -> rows r / r+8) ----
#pragma unroll
    for (int ms = 0; ms < 2; ++ms) {
#pragma unroll
        for (int ns = 0; ns < 4; ++ns) {
            const int n  = nBase + waveN + ns * 16 + laneN;
            const float bv = bias[n];
            const int rowB = mBase + waveM + ms * 16 + laneHi * 8;
#pragma unroll
            for (int r = 0; r < 8; ++r)
                out[(size_t)(rowB + r) * HID_ + n] = acc[ms][ns][r] + bv;
        }
    }
}

extern "C" void kernel_launch(void* const* d_in, const int* in_sizes, int n_in,
                              void* d_out, int out_size, void* d_ws, size_t ws_size,
                              hipStream_t stream) {
    const float* x    = (const float*)d_in[0];
    const float* w    = (const float*)d_in[1];
    const float* bias = (const float*)d_in[2];
    float*       out  = (float*)d_out;

    const size_t wtBytes = (size_t)KDIM_ * HID_ * sizeof(float);
    const int useWt = (ws_size >= wtBytes) ? 1 : 0;
    float* wt = (float*)d_ws;

    if (useWt) {
        wtrans_kernel<<<dim3(KDIM_ / 32, HID_ / 32), dim3(32, 8), 0, stream>>>(w, wt);
    }
    softsplit_gemm<<<dim3(PIX_ / 64, HID_ / 256), dim3(256), 0, stream>>>(
        x, w, wt, bias, out, useWt);
}